// WaveNet_29695403885259
// MI455X (gfx1250) — compile-verified
//
#include <hip/hip_runtime.h>
#include <math.h>

// ---------------------------------------------------------------------------
// WaveNet forward for MI455X (gfx1250).
//  - fp32 matmuls via V_WMMA_F32_16X16X4_F32
//  - per-block activation tiles staged to LDS by the Tensor Data Mover
//    (tensor_load_to_lds + s_wait_tensorcnt), padded to a 72-float row
//    stride for conflict-free ds reads.
// Workspace layout (floats): before0 | before1 | after0 | after1 | skip
// (5 x 4*128*16384*4B = 160 MB).
// ---------------------------------------------------------------------------

typedef float v2f __attribute__((ext_vector_type(2)));
typedef float v8f __attribute__((ext_vector_type(8)));
typedef unsigned int v4u __attribute__((ext_vector_type(4)));
typedef int v4i __attribute__((ext_vector_type(4)));
typedef int v8i __attribute__((ext_vector_type(8)));

#define LBUF 16384
#define LM1  16382      // last valid p (before/after length 16383)
#define RC   128
#define MC   80
#define CIN_ 256
#define LFIN 15360
#define SKIP_OFF 1023
#define LDSS 72         // LDS row stride (floats): 64 data + 8 pad

__device__ __forceinline__ v8f wmma4(v2f a, v2f b, v8f c) {
  // D = A(16x4) * B(4x16) + C, fp32
  return __builtin_amdgcn_wmma_f32_16x16x4_f32(false, a, false, b,
                                               (short)0, c, false, false);
}

__device__ __forceinline__ unsigned lds_off(const void* p) {
  // generic pointer into LDS aperture: low 32 bits are the LDS byte offset
  return (unsigned)(unsigned long long)p;
}

// 2D TDM load: rows x 64 cols tile, global row stride = row_stride elements,
// LDS rows padded to LDSS floats (pad_interval = 64 DW, pad_amount = 8 DW).
// rem_cols = columns remaining in the tensor from the tile start (OOB -> 0).
__device__ __forceinline__ void tdm_load_2d(const float* gptr,
                                            unsigned ldsoff,
                                            unsigned rem_cols,
                                            unsigned rows,
                                            unsigned row_stride)
{
  unsigned long long ga = (unsigned long long)gptr;
  v4u g0;
  g0[0] = 1u;                                   // count=1, user mode
  g0[1] = ldsoff;                               // lds_addr (bytes)
  g0[2] = (unsigned)(ga & 0xffffffffull);       // global_addr[31:0]
  g0[3] = (unsigned)((ga >> 32) & 0x1ffffffull) // global_addr[56:32]
        | 0x80000000u;                          // type = 2 ("image")
  v8i g1;
  // workgroup_mask=0 | data_size=4B(2)<<16 | pad_enable<<20 |
  // pad_interval=5 (64 DW) <<22 | pad_amount=7 (8 DW) <<25
  g1[0] = (2 << 16) | (1 << 20) | (5 << 22) | (7 << 25);
  g1[1] = (int)((rem_cols & 0xffffu) << 16);    // tensor_dim0[15:0] @bits63:48
  g1[2] = (int)((rem_cols >> 16) & 0xffffu)     // tensor_dim0[31:16]
        | (int)((rows & 0xffffu) << 16);        // tensor_dim1[15:0]
  g1[3] = (int)(64u << 16);                     // tensor_dim1[31:16]=0 | tile_dim0=64
  g1[4] = (int)rows;                            // tile_dim1=rows, tile_dim2=0
  g1[5] = (int)row_stride;                      // tensor_dim0_stride[31:0]
  g1[6] = 0;                                    // stride[47:32]=0 | dim1_stride lo
  g1[7] = 0;                                    // dim1_stride hi (unused, 2D)
  v4i gz4 = { 0, 0, 0, 0 };                     // groups 2/3: zero (<=2D tensor)
  v8i gz8 = { 0, 0, 0, 0, 0, 0, 0, 0 };         // extra group (clang-23 form): zero
  __builtin_amdgcn_tensor_load_to_lds(g0, g1, gz4, gz4, gz8, 0);
}

// ---------------------------------------------------------------------------
// Start convs: out[m][t] = sum_i W[m][i][0]*x[i][t] + W[m][i][1]*x[i][t+1] + b
// grid: (256 tiles, 4 batches, 2 tensors{before,after}); 256 threads.
// ---------------------------------------------------------------------------
__global__ __launch_bounds__(256) void start_kernel(
    const float* __restrict__ xb, const float* __restrict__ xa,
    const float* __restrict__ wb, const float* __restrict__ bb,
    const float* __restrict__ wa, const float* __restrict__ ba,
    float* __restrict__ ob, float* __restrict__ oa)
{
  const int lane = threadIdx.x & 31;
  const int wave = threadIdx.x >> 5;
  const int half = lane >> 4, r = lane & 15;
  const int b = blockIdx.y;
  const int p0 = 64 * blockIdx.x;

  const float* x  = (blockIdx.z == 0) ? xb : xa;
  const float* w  = (blockIdx.z == 0) ? wb : wa;
  const float* bs = (blockIdx.z == 0) ? bb : ba;
  float*       o  = (blockIdx.z == 0) ? ob : oa;
  x += (size_t)b * CIN_ * LBUF;
  o += (size_t)b * RC * LBUF;

  const int mrow = 16 * wave + r;
  v8f acc[4];
  {
    v8f ini;
#pragma unroll
    for (int j = 0; j < 8; ++j) ini[j] = bs[16 * wave + j + 8 * half];
    acc[0] = ini; acc[1] = ini; acc[2] = ini; acc[3] = ini;
  }
  int col[4];
#pragma unroll
  for (int s = 0; s < 4; ++s) {
    int c = p0 + 16 * s + r;
    col[s] = c > LM1 ? LM1 : c;
  }

  for (int k0 = 0; k0 < CIN_; k0 += 4) {
    const int kk = k0 + 2 * half;
    const float* wp = w + ((size_t)mrow * CIN_ + kk) * 2;   // [m][i][tap]
    v2f wl = *(const v2f*)(wp);
    v2f wh = *(const v2f*)(wp + 2);
    v2f a0 = { wl.x, wh.x };   // tap 0 over K=kk,kk+1
    v2f a1 = { wl.y, wh.y };   // tap 1
    const float* xr = x + (size_t)kk * LBUF;
#pragma unroll
    for (int s = 0; s < 4; ++s) {
      v2f f;
      f.x = xr[col[s]];            f.y = xr[LBUF + col[s]];
      acc[s] = wmma4(a0, f, acc[s]);
      f.x = xr[col[s] + 1];        f.y = xr[LBUF + col[s] + 1];
      acc[s] = wmma4(a1, f, acc[s]);
    }
  }

#pragma unroll
  for (int s = 0; s < 4; ++s) {
    int c = p0 + 16 * s + r;
    if (c <= LM1) {
#pragma unroll
      for (int j = 0; j < 8; ++j) {
        int m = 16 * wave + j + 8 * half;
        o[(size_t)m * LBUF + c] = acc[s][j];
      }
    }
  }
}

// GEMM with B operand held in LDS (row stride LDSS): acc = W(128x128)@B + bias
__device__ __forceinline__ void gemm_lds(const float* __restrict__ gl,
                                         const float* __restrict__ w,
                                         const float* __restrict__ bias,
                                         int wave, int half, int r, v8f acc[4])
{
  v8f ini;
#pragma unroll
  for (int j = 0; j < 8; ++j) ini[j] = bias[16 * wave + j + 8 * half];
  acc[0] = ini; acc[1] = ini; acc[2] = ini; acc[3] = ini;
  const int mrow = 16 * wave + r;
  for (int k0 = 0; k0 < RC; k0 += 4) {
    const int kk = k0 + 2 * half;
    v2f a = *(const v2f*)(w + (size_t)mrow * RC + kk);  // W[m][kk],W[m][kk+1]
#pragma unroll
    for (int s = 0; s < 4; ++s) {
      v2f f;
      f.x = gl[(size_t)kk * LDSS + 16 * s + r];
      f.y = gl[(size_t)(kk + 1) * LDSS + 16 * s + r];
      acc[s] = wmma4(a, f, acc[s]);
    }
  }
}

// ---------------------------------------------------------------------------
// One WaveNet block. grid: (tiles, 4 batches); 256 threads.
// Dynamic LDS: 4 tiles (128xLDSS) + 2 mel tiles (80xLDSS) + g (128xLDSS)
//            = 57600 floats = 225 KB per workgroup (one WG per WGP).
// ---------------------------------------------------------------------------
__global__ __launch_bounds__(256) void block_kernel(
    const float* __restrict__ bin, float* __restrict__ bout,
    const float* __restrict__ ain, float* __restrict__ aout,
    const float* __restrict__ mel,
    const float* __restrict__ wdb, const float* __restrict__ bdb,
    const float* __restrict__ wda, const float* __restrict__ bda,
    const float* __restrict__ wdm, const float* __restrict__ bdm,
    const float* __restrict__ wrb, const float* __restrict__ brb,
    const float* __restrict__ wra, const float* __restrict__ bra,
    const float* __restrict__ wsk, const float* __restrict__ bsk,
    float* __restrict__ skip, int d, int p_start, int first)
{
  extern __shared__ float smem[];
  float* bt0 = smem;                 // before @ p-d   [128 x LDSS]
  float* bt1 = bt0 + RC * LDSS;      // before @ p
  float* at0 = bt1 + RC * LDSS;      // after  @ p-d
  float* at1 = at0 + RC * LDSS;      // after  @ p
  float* mt0 = at1 + RC * LDSS;      // mel    @ p-d+1 [80 x LDSS]
  float* mt1 = mt0 + MC * LDSS;      // mel    @ p+1
  float* gld = mt1 + MC * LDSS;      // gated activation g [128 x LDSS]

  const int lane = threadIdx.x & 31;
  const int wave = threadIdx.x >> 5;
  const int half = lane >> 4, r = lane & 15;
  const int b = blockIdx.y;
  const int p0 = p_start + 64 * blockIdx.x;

  const float* binb = bin + (size_t)b * RC * LBUF;
  const float* ainb = ain + (size_t)b * RC * LBUF;
  const float* melb = mel + (size_t)b * MC * LBUF;

  // ---- TDM: stage the six activation tiles into LDS (wave 0 issues) ----
  if (wave == 0) {
    tdm_load_2d(binb + (p0 - d),     lds_off(bt0), 16383u - (p0 - d), RC, LBUF);
    tdm_load_2d(binb + p0,           lds_off(bt1), 16383u - p0,       RC, LBUF);
    tdm_load_2d(ainb + (p0 - d),     lds_off(at0), 16383u - (p0 - d), RC, LBUF);
    tdm_load_2d(ainb + p0,           lds_off(at1), 16383u - p0,       RC, LBUF);
    tdm_load_2d(melb + (p0 - d + 1), lds_off(mt0), 16384u - (p0 - d + 1), MC, LBUF);
    tdm_load_2d(melb + (p0 + 1),     lds_off(mt1), 16384u - (p0 + 1), MC, LBUF);
    __builtin_amdgcn_s_wait_tensorcnt(0);
  }
  // warm the weight path while the DMA runs
  __builtin_prefetch(wdb, 0, 1);
  __builtin_prefetch(wda, 0, 1);
  __builtin_prefetch(wdm, 0, 1);
  __builtin_prefetch(wrb, 0, 1);
  __builtin_prefetch(wra, 0, 1);
  __builtin_prefetch(wsk, 0, 1);
  __syncthreads();

  const int mrow = 16 * wave + r;
  const int cs = r;  // column-in-tile base (per subtile add 16*s)

  // ---------------- z accumulation (6 dilated-conv GEMMs) ----------------
  v8f z[4];
  {
    v8f ini;
#pragma unroll
    for (int j = 0; j < 8; ++j) {
      int m = 16 * wave + j + 8 * half;
      ini[j] = bdb[m] + bda[m] + bdm[m];
    }
    z[0] = ini; z[1] = ini; z[2] = ini; z[3] = ini;
  }

  for (int k0 = 0; k0 < RC; k0 += 4) {
    const int kk = k0 + 2 * half;

    const float* wp = wdb + ((size_t)mrow * RC + kk) * 2;
    v2f wl = *(const v2f*)wp, wh = *(const v2f*)(wp + 2);
    v2f ab0 = { wl.x, wh.x }, ab1 = { wl.y, wh.y };
    wp = wda + ((size_t)mrow * RC + kk) * 2;
    wl = *(const v2f*)wp; wh = *(const v2f*)(wp + 2);
    v2f aa0 = { wl.x, wh.x }, aa1 = { wl.y, wh.y };

    const float* tb0 = bt0 + (size_t)kk * LDSS;
    const float* tb1 = bt1 + (size_t)kk * LDSS;
    const float* ta0 = at0 + (size_t)kk * LDSS;
    const float* ta1 = at1 + (size_t)kk * LDSS;
#pragma unroll
    for (int s = 0; s < 4; ++s) {
      const int c = cs + 16 * s;
      v2f f;
      f.x = tb0[c]; f.y = tb0[LDSS + c];
      z[s] = wmma4(ab0, f, z[s]);
      f.x = tb1[c]; f.y = tb1[LDSS + c];
      z[s] = wmma4(ab1, f, z[s]);
      f.x = ta0[c]; f.y = ta0[LDSS + c];
      z[s] = wmma4(aa0, f, z[s]);
      f.x = ta1[c]; f.y = ta1[LDSS + c];
      z[s] = wmma4(aa1, f, z[s]);
    }

    if (k0 < MC) {   // mel contribution, K=80
      wp = wdm + ((size_t)mrow * MC + kk) * 2;
      wl = *(const v2f*)wp; wh = *(const v2f*)(wp + 2);
      v2f am0 = { wl.x, wh.x }, am1 = { wl.y, wh.y };
      const float* tm0 = mt0 + (size_t)kk * LDSS;
      const float* tm1 = mt1 + (size_t)kk * LDSS;
#pragma unroll
      for (int s = 0; s < 4; ++s) {
        const int c = cs + 16 * s;
        v2f f;
        f.x = tm0[c]; f.y = tm0[LDSS + c];
        z[s] = wmma4(am0, f, z[s]);
        f.x = tm1[c]; f.y = tm1[LDSS + c];
        z[s] = wmma4(am1, f, z[s]);
      }
    }
  }

  // ---------------- gated activation -> LDS ----------------
#pragma unroll
  for (int s = 0; s < 4; ++s) {
#pragma unroll
    for (int j = 0; j < 8; ++j) {
      float zz = z[s][j];
      float g  = tanhf(zz) * (1.0f / (1.0f + __expf(-zz)));
      int cch = 16 * wave + j + 8 * half;
      gld[(size_t)cch * LDSS + 16 * s + r] = g;
    }
  }
  __syncthreads();

  // ---------------- res_b ----------------
  v8f acc[4];
  gemm_lds(gld, wrb, brb, wave, half, r, acc);
  {
    float* boutb = bout + (size_t)b * RC * LBUF;
#pragma unroll
    for (int s = 0; s < 4; ++s) {
      int c = p0 + 16 * s + r;
      if (c <= LM1) {
#pragma unroll
        for (int j = 0; j < 8; ++j) {
          int m = 16 * wave + j + 8 * half;
          boutb[(size_t)m * LBUF + c] = acc[s][j] + binb[(size_t)m * LBUF + c];
        }
      }
    }
  }

  // ---------------- res_a ----------------
  gemm_lds(gld, wra, bra, wave, half, r, acc);
  {
    float* aoutb = aout + (size_t)b * RC * LBUF;
#pragma unroll
    for (int s = 0; s < 4; ++s) {
      int c = p0 + 16 * s + r;
      if (c <= LM1) {
#pragma unroll
        for (int j = 0; j < 8; ++j) {
          int m = 16 * wave + j + 8 * half;
          aoutb[(size_t)m * LBUF + c] = acc[s][j] + ainb[(size_t)m * LBUF + c];
        }
      }
    }
  }

  // ---------------- skip accumulate ----------------
  gemm_lds(gld, wsk, bsk, wave, half, r, acc);
  {
    float* skb = skip + (size_t)b * RC * LBUF;
#pragma unroll
    for (int s = 0; s < 4; ++s) {
      int c = p0 + 16 * s + r;
      if (c >= SKIP_OFF && c <= LM1) {
#pragma unroll
        for (int j = 0; j < 8; ++j) {
          int m = 16 * wave + j + 8 * half;
          size_t idx = (size_t)m * LBUF + c;
          float v = acc[s][j];
          if (first) skb[idx] = v;
          else       skb[idx] = skb[idx] + v;
        }
      }
    }
  }
}

// ---------------------------------------------------------------------------
// End: relu(skip) -> relu(W1@. + b1) -> W2@. + b2  -> out [4,256,15360]
// grid: (240 tiles, 4 batches)
// ---------------------------------------------------------------------------
__global__ __launch_bounds__(256) void end_kernel(
    const float* __restrict__ skip,
    const float* __restrict__ w1, const float* __restrict__ b1,
    const float* __restrict__ w2, const float* __restrict__ b2,
    float* __restrict__ out)
{
  __shared__ float y_lds[RC * LDSS];

  const int lane = threadIdx.x & 31;
  const int wave = threadIdx.x >> 5;
  const int half = lane >> 4, r = lane & 15;
  const int b = blockIdx.y;
  const int s0 = 64 * blockIdx.x;

  const float* skb = skip + (size_t)b * RC * LBUF + SKIP_OFF;
  const int mrow = 16 * wave + r;

  // stage 1: y1 = relu(W1 @ relu(skip) + b1)
  v8f acc[4];
  {
    v8f ini;
#pragma unroll
    for (int j = 0; j < 8; ++j) ini[j] = b1[16 * wave + j + 8 * half];
    acc[0] = ini; acc[1] = ini; acc[2] = ini; acc[3] = ini;
  }
  for (int k0 = 0; k0 < RC; k0 += 4) {
    const int kk = k0 + 2 * half;
    v2f a = *(const v2f*)(w1 + (size_t)mrow * RC + kk);
#pragma unroll
    for (int s = 0; s < 4; ++s) {
      int cs = s0 + 16 * s + r;
      v2f f;
      f.x = fmaxf(0.0f, skb[(size_t)kk * LBUF + cs]);
      f.y = fmaxf(0.0f, skb[(size_t)(kk + 1) * LBUF + cs]);
      acc[s] = wmma4(a, f, acc[s]);
    }
  }
#pragma unroll
  for (int s = 0; s < 4; ++s) {
#pragma unroll
    for (int j = 0; j < 8; ++j) {
      int cch = 16 * wave + j + 8 * half;
      y_lds[(size_t)cch * LDSS + 16 * s + r] = fmaxf(0.0f, acc[s][j]);
    }
  }
  __syncthreads();

  // stage 2: out = W2(256x128) @ y1 + b2 ; two 16-row slabs per wave
  float* ob = out + (size_t)b * CIN_ * LFIN;
#pragma unroll
  for (int slab = 0; slab < 2; ++slab) {
    const int mbase = 128 * slab;
    v8f a2[4];
    {
      v8f ini;
#pragma unroll
      for (int j = 0; j < 8; ++j) ini[j] = b2[mbase + 16 * wave + j + 8 * half];
      a2[0] = ini; a2[1] = ini; a2[2] = ini; a2[3] = ini;
    }
    const int m2 = mbase + mrow;
    for (int k0 = 0; k0 < RC; k0 += 4) {
      const int kk = k0 + 2 * half;
      v2f a = *(const v2f*)(w2 + (size_t)m2 * RC + kk);
#pragma unroll
      for (int s = 0; s < 4; ++s) {
        v2f f;
        f.x = y_lds[(size_t)kk * LDSS + 16 * s + r];
        f.y = y_lds[(size_t)(kk + 1) * LDSS + 16 * s + r];
        a2[s] = wmma4(a, f, a2[s]);
      }
    }
#pragma unroll
    for (int s = 0; s < 4; ++s) {
      int cs = s0 + 16 * s + r;
#pragma unroll
      for (int j = 0; j < 8; ++j) {
        int m = mbase + 16 * wave + j + 8 * half;
        ob[(size_t)m * LFIN + cs] = a2[s][j];
      }
    }
  }
}

// ---------------------------------------------------------------------------
extern "C" void kernel_launch(void* const* d_in, const int* in_sizes, int n_in,
                              void* d_out, int out_size, void* d_ws, size_t ws_size,
                              hipStream_t stream)
{
  const float* before    = (const float*)d_in[0];
  const float* after     = (const float*)d_in[1];
  const float* mel       = (const float*)d_in[2];
  const float* w_start_b = (const float*)d_in[3];
  const float* b_start_b = (const float*)d_in[4];
  const float* w_start_a = (const float*)d_in[5];
  const float* b_start_a = (const float*)d_in[6];
  const float* w_dil_b   = (const float*)d_in[7];
  const float* b_dil_b   = (const float*)d_in[8];
  const float* w_dil_a   = (const float*)d_in[9];
  const float* b_dil_a   = (const float*)d_in[10];
  const float* w_dil_m   = (const float*)d_in[11];
  const float* b_dil_m   = (const float*)d_in[12];
  const float* w_res_b   = (const float*)d_in[13];
  const float* b_res_b   = (const float*)d_in[14];
  const float* w_res_a   = (const float*)d_in[15];
  const float* b_res_a   = (const float*)d_in[16];
  const float* w_skip    = (const float*)d_in[17];
  const float* b_skip    = (const float*)d_in[18];
  const float* w_end1    = (const float*)d_in[19];
  const float* b_end1    = (const float*)d_in[20];
  const float* w_end2    = (const float*)d_in[21];
  const float* b_end2    = (const float*)d_in[22];

  const size_t BUF = (size_t)4 * RC * LBUF;      // floats per activation buffer
  float* ws = (float*)d_ws;
  float* bbuf[2] = { ws,           ws + BUF };
  float* abuf[2] = { ws + 2 * BUF, ws + 3 * BUF };
  float* skip    = ws + 4 * BUF;                 // needs ws_size >= 160 MB

  dim3 blk(256);

  start_kernel<<<dim3(256, 4, 2), blk, 0, stream>>>(
      before, after, w_start_b, b_start_b, w_start_a, b_start_a,
      bbuf[0], abuf[0]);

  // dynamic LDS: 4*(128*LDSS) + 2*(80*LDSS) + 128*LDSS floats
  const size_t shmem = (size_t)(4 * RC + 2 * MC + RC) * LDSS * sizeof(float);

  for (int i = 0; i < 10; ++i) {
    const int dd = 1 << i;
    const int p_start = 2 * dd - 1;          // 2^(i+1) - 1
    const int ncols = LBUF - 2 * dd;         // 16384 - 2^(i+1)
    const int tiles = (ncols + 63) / 64;
    block_kernel<<<dim3(tiles, 4), blk, shmem, stream>>>(
        bbuf[i & 1], bbuf[(i + 1) & 1],
        abuf[i & 1], abuf[(i + 1) & 1], mel,
        w_dil_b + (size_t)i * RC * RC * 2, b_dil_b + (size_t)i * RC,
        w_dil_a + (size_t)i * RC * RC * 2, b_dil_a + (size_t)i * RC,
        w_dil_m + (size_t)i * RC * MC * 2, b_dil_m + (size_t)i * RC,
        w_res_b + (size_t)i * RC * RC,     b_res_b + (size_t)i * RC,
        w_res_a + (size_t)i * RC * RC,     b_res_a + (size_t)i * RC,
        w_skip  + (size_t)i * RC * RC,     b_skip  + (size_t)i * RC,
        skip, dd, p_start, i == 0 ? 1 : 0);
  }

  end_kernel<<<dim3(LFIN / 64, 4), blk, 0, stream>>>(
      skip, w_end1, b_end1, w_end2, b_end2, (float*)d_out);
}